// PDE_GCN_32444182954622
// MI455X (gfx1250) — compile-verified
//
#include <hip/hip_runtime.h>
#include <hip/hip_bf16.h>
#include <math.h>

typedef __attribute__((ext_vector_type(2))) float v2f;
typedef __attribute__((ext_vector_type(8))) float v8f;

constexpr int kN = 6;     // NNODES
constexpr int kC = 256;   // channels

__device__ __forceinline__ constexpr int tri(int n, int m) {
  // index into packed upper triangle (n<=m), 21 entries for kN=6
  return n * kN - n * (n - 1) / 2 + (m - n);
}

// Native CDNA5 V_TANH_F32 when available (gfx1250 tanh-insts); OCML fallback.
__device__ __forceinline__ float fast_tanh(float x) {
#if __has_builtin(__builtin_amdgcn_tanhf)
  return __builtin_amdgcn_tanhf(x);
#elif __has_builtin(__builtin_amdgcn_tanh_f32)
  return __builtin_amdgcn_tanh_f32(x);
#else
  return tanhf(x);
#endif
}

// ---------------------------------------------------------------------------
// Kernel 1: opening GEMM  xn0[r][o] = relu(sum_c x[r][c] * K1[o][c])
//   r = b*6+n (M=49152), o (N=256), c (K=256); fp32 WMMA 16x16x4.
//   One 16x16 output tile per wave per col-step; 8 waves/block cover 128 rows.
// ---------------------------------------------------------------------------
__global__ __launch_bounds__(256, 1)
void open_gemm_wmma(const float* __restrict__ x, const float* __restrict__ K1,
                    float* __restrict__ xn0, int rows) {
  const int lane = threadIdx.x & 31;
  const int wave = threadIdx.x >> 5;
  const int rowbase = (blockIdx.x * 8 + wave) * 16;
  if (rowbase + 16 > rows) return;                 // wave-uniform: EXEC stays all-1s
  const int half = lane >> 4;                      // 0: K lo pair, 1: K hi pair
  const int l16  = lane & 15;

  // A fragment source: lane (l16, half) covers A[M=l16][k + 2*half + {0,1}]
  const float* aptr = x + (size_t)(rowbase + l16) * kC + 2 * half;

  for (int ct = 0; ct < kC / 16; ++ct) {
    const int colbase = ct * 16;
    // B[k][n] = K1[n][k]; lane covers B[k + 2*half + {0,1}][n=l16]
    const float* bptr = K1 + (size_t)(colbase + l16) * kC + 2 * half;
    v8f acc = {};
#pragma unroll 8
    for (int k = 0; k < kC; k += 4) {
      v2f a = *(const v2f*)(aptr + k);
      v2f b = *(const v2f*)(bptr + k);
      acc = __builtin_amdgcn_wmma_f32_16x16x4_f32(
          /*neg_a=*/false, a, /*neg_b=*/false, b,
          /*c_mod=*/(short)0, acc, /*reuse_a=*/false, /*reuse_b=*/false);
    }
    // D layout: VGPR r -> M = r + 8*half, N = l16
#pragma unroll
    for (int r = 0; r < 8; ++r) {
      const int row = rowbase + r + 8 * half;
      const int col = colbase + l16;
      xn0[(size_t)row * kC + col] = fmaxf(acc[r], 0.0f);
    }
  }
}

// ---------------------------------------------------------------------------
// Kernel 2: fused iterative PDE, one wave per batch.
//   Lane l owns channels c = l + 32*j, j=0..7.  xn/xn_old live in VGPRs.
//   Gram + closing projection reduced across the wave with __shfl_xor.
// ---------------------------------------------------------------------------
__global__ __launch_bounds__(256, 1)
void pde_iter(const float* __restrict__ xn0, const float* __restrict__ KN,
              const float* __restrict__ alpha_p, const float* __restrict__ h_p,
              float* __restrict__ out0, float* __restrict__ xn_out,
              float* __restrict__ wl_out, int B) {
  __shared__ float sbuf[8 * kC * kN];  // 48 KB: per-wave staging for coalesced xn store
  const int lane = threadIdx.x & 31;
  const int wave = threadIdx.x >> 5;
  const int b = blockIdx.x * 8 + wave;
  if (b >= B) return;

  const float alpha = alpha_p[0];
  const float h = h_p[0];
  const float beta = 1.0f / (1.0f + expf(-alpha));
  const float al = (1.0f - beta) / h;
  const float be = beta / (h * h);
  const float inv_den = 1.0f / (be + al);
  const float c1 = 2.0f * be + al;

  float xn[8][kN], xo[8][kN];
#pragma unroll
  for (int j = 0; j < 8; ++j) {
    const int c = lane + 32 * j;
#pragma unroll
    for (int n = 0; n < kN; ++n) {
      xn[j][n] = xn0[((size_t)b * kN + n) * kC + c];   // coalesced 128B per (j,n)
      xo[j][n] = xn[j][n];
    }
  }

  for (int it = 0; it < 4; ++it) {
    // ---- per-lane partial Gram over owned channels; 21 unique pairs ----
    float G[21];
#pragma unroll
    for (int n = 0; n < kN; ++n)
#pragma unroll
      for (int m = n; m < kN; ++m) {
        float s = 0.0f;
#pragma unroll
        for (int j = 0; j < 8; ++j) s = fmaf(xn[j][n], xn[j][m], s);
        G[tri(n, m)] = s;
      }
#pragma unroll
    for (int off = 16; off >= 1; off >>= 1)
#pragma unroll
      for (int i = 0; i < 21; ++i) G[i] += __shfl_xor(G[i], off, 32);

    // ---- D = relu(pairwise sq-dist); std (ddof=1); W1 = exp(-2D/std)-10I ----
    float D[kN * kN];
#pragma unroll
    for (int n = 0; n < kN; ++n)
#pragma unroll
      for (int m = 0; m < kN; ++m) {
        if (n == m) { D[n * kN + m] = 0.0f; continue; }
        const int lo = n < m ? n : m, hi = n < m ? m : n;
        const float d = G[tri(n, n)] + G[tri(m, m)] - 2.0f * G[tri(lo, hi)];
        D[n * kN + m] = fmaxf(d, 0.0f);
      }
    float mean = 0.0f;
#pragma unroll
    for (int i = 0; i < 36; ++i) mean += D[i];
    mean *= (1.0f / 36.0f);
    float var = 0.0f;
#pragma unroll
    for (int i = 0; i < 36; ++i) { const float e = D[i] - mean; var = fmaf(e, e, var); }
    const float sinv = -2.0f / sqrtf(var * (1.0f / 35.0f));
    float W1[kN * kN];
#pragma unroll
    for (int n = 0; n < kN; ++n)
#pragma unroll
      for (int m = 0; m < kN; ++m)
        W1[n * kN + m] = expf(D[n * kN + m] * sinv) - (n == m ? 10.0f : 0.0f);

    // ---- edge grad -> tanh^5 -> edge div -> implicit update ----
#pragma unroll
    for (int j = 0; j < 8; ++j) {
      float nx[kN];
#pragma unroll
      for (int m = 0; m < kN; ++m) {
        float s = 0.0f;
#pragma unroll
        for (int n = 0; n < kN; ++n) {
          const float w = W1[n * kN + m];
          float t = w * (xn[j][m] - xn[j][n]);
          t = fast_tanh(t); t = fast_tanh(t); t = fast_tanh(t);
          t = fast_tanh(t); t = fast_tanh(t);
          s = fmaf(w, t, s);
        }
        nx[m] = (c1 * xn[j][m] - be * xo[j][m] - s) * inv_den;
      }
#pragma unroll
      for (int m = 0; m < kN; ++m) { xo[j][m] = xn[j][m]; xn[j][m] = nx[m]; }
    }
  }

  // ---- closing projection T[o][n] = sum_c KNclose[o][c] * xn[c][n] ----
  float T[kN * kN];
#pragma unroll
  for (int i = 0; i < 36; ++i) T[i] = 0.0f;
#pragma unroll
  for (int j = 0; j < 8; ++j) {
    const int c = lane + 32 * j;
#pragma unroll
    for (int o = 0; o < kN; ++o) {
      const float k8 = KN[(size_t)o * kC + c];
#pragma unroll
      for (int n = 0; n < kN; ++n) T[o * kN + n] = fmaf(k8, xn[j][n], T[o * kN + n]);
    }
  }
#pragma unroll
  for (int off = 16; off >= 1; off >>= 1)
#pragma unroll
    for (int i = 0; i < 36; ++i) T[i] += __shfl_xor(T[i], off, 32);

  // out[:,0,:]  (out[b][0][o] = T[o][0])
  if (lane < kN) out0[(size_t)b * kN + lane] = T[lane * kN + 0];

  // final xn [b][c][n]: stage in LDS, then stream out coalesced (same-wave DS is in-order)
  float* sw = sbuf + wave * (kC * kN);
#pragma unroll
  for (int j = 0; j < 8; ++j) {
    const int c = lane + 32 * j;
#pragma unroll
    for (int n = 0; n < kN; ++n) sw[c * kN + n] = xn[j][n];
  }
#pragma unroll 4
  for (int t = 0; t < (kC * kN) / 32; ++t) {
    // write-once stream: nontemporal so it doesn't evict L2-resident K1/xn0
    __builtin_nontemporal_store(sw[t * 32 + lane],
                                &xn_out[(size_t)b * (kC * kN) + t * 32 + lane]);
  }

  // ---- Wl = _compute_W(out): rows of T are the feature rows ----
  float sql[kN];
#pragma unroll
  for (int i = 0; i < kN; ++i) {
    float s = 0.0f;
#pragma unroll
    for (int j = 0; j < kN; ++j) s = fmaf(T[i * kN + j], T[i * kN + j], s);
    sql[i] = s;
  }
  float Dl[kN * kN];
#pragma unroll
  for (int i = 0; i < kN; ++i)
#pragma unroll
    for (int k = 0; k < kN; ++k) {
      if (i == k) { Dl[i * kN + k] = 0.0f; continue; }
      float dot = 0.0f;
#pragma unroll
      for (int j = 0; j < kN; ++j) dot = fmaf(T[i * kN + j], T[k * kN + j], dot);
      Dl[i * kN + k] = fmaxf(sql[i] + sql[k] - 2.0f * dot, 0.0f);
    }
  float meanl = 0.0f;
#pragma unroll
  for (int i = 0; i < 36; ++i) meanl += Dl[i];
  meanl *= (1.0f / 36.0f);
  float varl = 0.0f;
#pragma unroll
  for (int i = 0; i < 36; ++i) { const float e = Dl[i] - meanl; varl = fmaf(e, e, varl); }
  const float sinvl = -2.0f / sqrtf(varl * (1.0f / 35.0f));

  // Wl flat layout: b*72 + n*12 + m*2 + comp ; comp0 = I, comp1 = exp(..)-10I
#pragma unroll
  for (int t = 0; t < 3; ++t) {
    const int f = lane + 32 * t;
    if (f < 72) {
      const int comp = f & 1;
      const int mm = (f >> 1) % kN;
      const int nn = f / 12;
      float val;
      if (comp == 0) val = (nn == mm) ? 1.0f : 0.0f;
      else           val = expf(Dl[nn * kN + mm] * sinvl) - ((nn == mm) ? 10.0f : 0.0f);
      wl_out[(size_t)b * 72 + f] = val;
    }
  }
}

// ---------------------------------------------------------------------------
extern "C" void kernel_launch(void* const* d_in, const int* in_sizes, int n_in,
                              void* d_out, int out_size, void* d_ws, size_t ws_size,
                              hipStream_t stream) {
  const float* x  = (const float*)d_in[0];   // [B, 6, 256]
  const float* K1 = (const float*)d_in[1];   // [256, 256]
  const float* KN = (const float*)d_in[2];   // [6, 256]
  const float* al = (const float*)d_in[3];   // [1,1]
  const float* hh = (const float*)d_in[4];   // [1]

  const int B = in_sizes[0] / (kN * kC);     // 8192
  float* xn0 = (float*)d_ws;                 // B*6*256 floats (~50 MB) scratch

  const int rows = B * kN;                   // 49152
  const int g1 = (rows + 127) / 128;         // 8 waves x 16 rows per block
  open_gemm_wmma<<<g1, 256, 0, stream>>>(x, K1, xn0, rows);

  float* out0   = (float*)d_out;
  float* xn_out = out0 + (size_t)B * kN;
  float* wl_out = xn_out + (size_t)B * kC * kN;
  const int g2 = (B + 7) / 8;                // one wave per batch
  pde_iter<<<g2, 256, 0, stream>>>(xn0, KN, al, hh, out0, xn_out, wl_out, B);
}